// MultiRelationGNN_22351009808444
// MI455X (gfx1250) — compile-verified
//
#include <hip/hip_runtime.h>

// ---------------------------------------------------------------------------
// MultiRelationGNN on MI455X (gfx1250, wave32).
//
// Algebraic restructure:
//   msg_e = w_e * ([h_src, h_dst] @ W_r + b_r)
//         = w_e * (h_src @ W_r_top) + w_e * (h_dst @ W_r_bot + b_r)
//   aggr[v] = sum_{e->v} w_e * S[src_e, r_e]                    (edge gather/scatter)
//           + sum_r Wsum[v,r] * (h[v] @ W_r_bot + b_r)          (node-level GEMM)
// where S[n,r] = h[n] @ W_r_top (dense GEMM, WMMA) and
// Wsum[v,r] = sum of w_e over edges of type r into v (scalar atomics).
// Wsum depends only on edge data -> computed once, reused by both layers.
//
// All dense math uses V_WMMA_F32_16X16X4_F32 (full f32 precision).
// ---------------------------------------------------------------------------

typedef __attribute__((ext_vector_type(2))) float v2f;
typedef __attribute__((ext_vector_type(8))) float v8f;

#define LRELU_SLOPE 0.01f

// --- WMMA f32 16x16x4 fragment loaders (row-major global memory) -----------
// A (16x4): lanes 0-15 hold M=lane, K=0..1 in VGPR0..1; lanes 16-31 hold K=2..3.
__device__ __forceinline__ v2f load_a_frag(const float* __restrict__ p, int ld,
                                           int m0, int k0, int lane) {
  int m = m0 + (lane & 15);
  int k = k0 + ((lane >> 4) << 1);
  const float* q = p + (size_t)m * ld + k;
  v2f a; a.x = q[0]; a.y = q[1];
  return a;
}
// B (4x16): VGPR v holds B[k0 + v + 2*half][n], n = lane&15.
__device__ __forceinline__ v2f load_b_frag(const float* __restrict__ p, int ld,
                                           int k0, int n0, int lane) {
  int n = n0 + (lane & 15);
  int k = k0 + ((lane >> 4) << 1);
  v2f b; b.x = p[(size_t)k * ld + n]; b.y = p[(size_t)(k + 1) * ld + n];
  return b;
}

// 16x16 tile of A[m0:,:64] @ B[:64, n0:], both row-major with given ld=64.
__device__ __forceinline__ v8f gemm64(const float* __restrict__ A,
                                      const float* __restrict__ B,
                                      int m0, int n0, int lane) {
  v8f acc = {};
#pragma unroll
  for (int k0 = 0; k0 < 64; k0 += 4) {
    v2f a = load_a_frag(A, 64, m0, k0, lane);
    v2f b = load_b_frag(B, 64, k0, n0, lane);
    acc = __builtin_amdgcn_wmma_f32_16x16x4_f32(false, a, false, b, (short)0,
                                                acc, false, false);
  }
  return acc;
}

// ---------------------------------------------------------------------------
__global__ void zero_kernel(float* __restrict__ p, size_t n) {
  size_t i = (size_t)blockIdx.x * blockDim.x + threadIdx.x;
  size_t stride = (size_t)gridDim.x * blockDim.x;
  for (; i < n; i += stride) p[i] = 0.0f;
}

// h0[N,64] = x[N,128] @ W_field[128,64] + b_field
__global__ void field_gemm_kernel(const float* __restrict__ x,
                                  const float* __restrict__ W,
                                  const float* __restrict__ bias,
                                  float* __restrict__ h0, int tilesM) {
  int wave = (int)(((size_t)blockIdx.x * blockDim.x + threadIdx.x) >> 5);
  int lane = threadIdx.x & 31;
  if (wave >= tilesM * 4) return;
  int m0 = (wave >> 2) * 16, n0 = (wave & 3) * 16;
  v8f acc = {};
#pragma unroll
  for (int k0 = 0; k0 < 128; k0 += 4) {
    v2f a = load_a_frag(x, 128, m0, k0, lane);
    v2f b = load_b_frag(W, 64, k0, n0, lane);
    acc = __builtin_amdgcn_wmma_f32_16x16x4_f32(false, a, false, b, (short)0,
                                                acc, false, false);
  }
  int n = n0 + (lane & 15);
  int half = lane >> 4;
  float bv = bias[n];
#pragma unroll
  for (int v = 0; v < 8; ++v) {
    int m = m0 + v + 8 * half;
    h0[(size_t)m * 64 + n] = acc[v] + bv;
  }
}

// Wsum[v,r] += lambda * exp(-beta * t_e) for every edge e of type r into v.
__global__ void edge_weight_kernel(const int* __restrict__ ei,
                                   const int* __restrict__ et,
                                   const float* __restrict__ etime,
                                   const float* __restrict__ lam,
                                   const float* __restrict__ beta,
                                   float* __restrict__ Wsum, int E) {
  int e = blockIdx.x * blockDim.x + threadIdx.x;
  if (e >= E) return;
  float w = lam[0] * __expf(-beta[0] * etime[e]);
  int dst = ei[E + e];
  atomicAdd(Wsum + (size_t)dst * 8 + et[e], w);
}

// S[n, r*64+j] = h[n,:] @ W_rel[r][0:64, j]   (src-side transform, all relations)
__global__ void src_transform_kernel(const float* __restrict__ h,
                                     const float* __restrict__ Wrel,
                                     float* __restrict__ S, int tilesM) {
  int wave = (int)(((size_t)blockIdx.x * blockDim.x + threadIdx.x) >> 5);
  int lane = threadIdx.x & 31;
  if (wave >= tilesM * 32) return;
  int tm = wave >> 5, tn = wave & 31;
  int r = tn >> 2, jb = tn & 3;
  int m0 = tm * 16, n0 = jb * 16;
  const float* B = Wrel + (size_t)r * 128 * 64;  // top half rows k=0..63
  v8f acc = gemm64(h, B, m0, n0, lane);
  int n = n0 + (lane & 15);
  int half = lane >> 4;
#pragma unroll
  for (int v = 0; v < 8; ++v) {
    int m = m0 + v + 8 * half;
    S[(size_t)m * 512 + r * 64 + n] = acc[v];
  }
}

// hout[dst] += w_e * S[src, r_e]   (one wave per edge, 2 floats per lane)
__global__ void edge_scatter_kernel(const int* __restrict__ ei,
                                    const int* __restrict__ et,
                                    const float* __restrict__ etime,
                                    const float* __restrict__ lam,
                                    const float* __restrict__ beta,
                                    const float* __restrict__ S,
                                    float* __restrict__ hout, int E) {
  size_t gid = (size_t)blockIdx.x * blockDim.x + threadIdx.x;
  int e = (int)(gid >> 5);
  int lane = threadIdx.x & 31;
  if (e >= E) return;
  int src = ei[e];
  int dst = ei[E + e];
  int r = et[e];
  float w = lam[0] * __expf(-beta[0] * etime[e]);
  const float2 sv =
      *(const float2*)(S + (size_t)src * 512 + r * 64 + lane * 2);
  float* orow = hout + (size_t)dst * 64 + lane * 2;
  atomicAdd(orow + 0, w * sv.x);
  atomicAdd(orow + 1, w * sv.y);
}

// hout[v] += sum_r Wsum[v,r] * (h[v] @ W_rel[r][64:128,:] + b_rel[r])
__global__ void dst_contrib_kernel(const float* __restrict__ h,
                                   const float* __restrict__ Wrel,
                                   const float* __restrict__ brel,
                                   const float* __restrict__ Wsum,
                                   float* __restrict__ hout, int tilesM) {
  int wave = (int)(((size_t)blockIdx.x * blockDim.x + threadIdx.x) >> 5);
  int lane = threadIdx.x & 31;
  if (wave >= tilesM * 4) return;
  int m0 = (wave >> 2) * 16, n0 = (wave & 3) * 16;
  int n = n0 + (lane & 15);
  int half = lane >> 4;
  float tot[8] = {0.f, 0.f, 0.f, 0.f, 0.f, 0.f, 0.f, 0.f};
  for (int r = 0; r < 8; ++r) {
    const float* B = Wrel + (size_t)r * 128 * 64 + 64 * 64;  // bottom half
    v8f acc = gemm64(h, B, m0, n0, lane);
    float bv = brel[r * 64 + n];
#pragma unroll
    for (int v = 0; v < 8; ++v) {
      int node = m0 + v + 8 * half;
      float s = Wsum[(size_t)node * 8 + r];
      tot[v] += s * (acc[v] + bv);
    }
  }
#pragma unroll
  for (int v = 0; v < 8; ++v) {
    int node = m0 + v + 8 * half;
    size_t idx = (size_t)node * 64 + n;
    hout[idx] += tot[v];  // exclusive tile ownership; runs after edge_scatter
  }
}

// out = lrelu(h2@W2+b2) + lrelu(h1@W1+b1) + lrelu(h0@W0+b0)
__global__ void out_head_kernel(const float* __restrict__ h0,
                                const float* __restrict__ h1,
                                const float* __restrict__ h2,
                                const float* __restrict__ W0,
                                const float* __restrict__ b0,
                                const float* __restrict__ W1,
                                const float* __restrict__ b1,
                                const float* __restrict__ W2,
                                const float* __restrict__ b2,
                                float* __restrict__ out, int tilesM) {
  int wave = (int)(((size_t)blockIdx.x * blockDim.x + threadIdx.x) >> 5);
  int lane = threadIdx.x & 31;
  if (wave >= tilesM * 4) return;
  int m0 = (wave >> 2) * 16, n0 = (wave & 3) * 16;
  int n = n0 + (lane & 15);
  int half = lane >> 4;
  float res[8] = {0.f, 0.f, 0.f, 0.f, 0.f, 0.f, 0.f, 0.f};

  const float* Hs[3] = {h0, h1, h2};
  const float* Ws[3] = {W0, W1, W2};
  const float* Bs[3] = {b0, b1, b2};
#pragma unroll
  for (int head = 0; head < 3; ++head) {
    v8f acc = gemm64(Hs[head], Ws[head], m0, n0, lane);
    float bv = Bs[head][n];
#pragma unroll
    for (int v = 0; v < 8; ++v) {
      float t = acc[v] + bv;
      res[v] += (t > 0.0f) ? t : LRELU_SLOPE * t;
    }
  }
#pragma unroll
  for (int v = 0; v < 8; ++v) {
    int m = m0 + v + 8 * half;
    out[(size_t)m * 64 + n] = res[v];
  }
}

// ---------------------------------------------------------------------------
extern "C" void kernel_launch(void* const* d_in, const int* in_sizes, int n_in,
                              void* d_out, int out_size, void* d_ws,
                              size_t ws_size, hipStream_t stream) {
  const float* x = (const float*)d_in[0];
  const int* ei = (const int*)d_in[1];
  const int* et = (const int*)d_in[2];
  const float* etime = (const float*)d_in[3];
  const float* lam = (const float*)d_in[4];
  const float* beta = (const float*)d_in[5];
  const float* W_field = (const float*)d_in[6];
  const float* b_field = (const float*)d_in[7];
  const float* W_rel1 = (const float*)d_in[8];
  const float* b_rel1 = (const float*)d_in[9];
  const float* W_rel2 = (const float*)d_in[10];
  const float* b_rel2 = (const float*)d_in[11];
  const float* W_out0 = (const float*)d_in[12];
  const float* b_out0 = (const float*)d_in[13];
  const float* W_out1 = (const float*)d_in[14];
  const float* b_out1 = (const float*)d_in[15];
  const float* W_out2 = (const float*)d_in[16];
  const float* b_out2 = (const float*)d_in[17];
  float* out = (float*)d_out;

  const int N = in_sizes[0] / 128;  // 50000
  const int E = in_sizes[2];        // 1600000
  const int tilesM = N / 16;        // 3125 (N is a multiple of 16)

  // Workspace layout (f32): h0|h1|h2 (N*64 each), S (N*512), Wsum (N*8)
  float* h0 = (float*)d_ws;
  float* h1 = h0 + (size_t)N * 64;
  float* h2 = h1 + (size_t)N * 64;
  float* S = h2 + (size_t)N * 64;
  float* Wsum = S + (size_t)N * 512;

  const int TPB = 256;
  auto cdiv = [](long long a, long long b) { return (int)((a + b - 1) / b); };

  int grid_tiles4 = cdiv((long long)tilesM * 4 * 32, TPB);
  int grid_tiles32 = cdiv((long long)tilesM * 32 * 32, TPB);

  // h0 = x @ W_field + b_field
  field_gemm_kernel<<<grid_tiles4, TPB, 0, stream>>>(x, W_field, b_field, h0,
                                                     tilesM);

  // Wsum depends only on edges: compute once, reuse for both layers.
  zero_kernel<<<1024, TPB, 0, stream>>>(Wsum, (size_t)N * 8);
  edge_weight_kernel<<<cdiv(E, TPB), TPB, 0, stream>>>(ei, et, etime, lam,
                                                       beta, Wsum, E);

  // ---- Layer 1: h0 -> h1 ----
  zero_kernel<<<4096, TPB, 0, stream>>>(h1, (size_t)N * 64);
  src_transform_kernel<<<grid_tiles32, TPB, 0, stream>>>(h0, W_rel1, S, tilesM);
  edge_scatter_kernel<<<cdiv((long long)E * 32, TPB), TPB, 0, stream>>>(
      ei, et, etime, lam, beta, S, h1, E);
  dst_contrib_kernel<<<grid_tiles4, TPB, 0, stream>>>(h0, W_rel1, b_rel1, Wsum,
                                                      h1, tilesM);

  // ---- Layer 2: h1 -> h2 ----
  zero_kernel<<<4096, TPB, 0, stream>>>(h2, (size_t)N * 64);
  src_transform_kernel<<<grid_tiles32, TPB, 0, stream>>>(h1, W_rel2, S, tilesM);
  edge_scatter_kernel<<<cdiv((long long)E * 32, TPB), TPB, 0, stream>>>(
      ei, et, etime, lam, beta, S, h2, E);
  dst_contrib_kernel<<<grid_tiles4, TPB, 0, stream>>>(h1, W_rel2, b_rel2, Wsum,
                                                      h2, tilesM);

  // ---- Output heads ----
  out_head_kernel<<<grid_tiles4, TPB, 0, stream>>>(
      h0, h1, h2, W_out0, b_out0, W_out1, b_out1, W_out2, b_out2, out, tilesM);
}